// frame_similarly_77988016161334
// MI455X (gfx1250) — compile-verified
//
#include <hip/hip_runtime.h>
#include <stdint.h>

// Problem: frames (N=16, B=4, C=64, H=128, W=128) f32.
// out[n, idx] = frames[n, idx]^2 * sum_m frames[m, idx]
// Memory bound: 256MB in + 256MB out @ 23.3 TB/s ~= 22us. FLOPs negligible.
// Strategy: 1 thread per 4 consecutive spatial f32 (b128 access). The 16
// frame values are staged into LDS with gfx1250 async global->LDS b128 loads
// (ASYNCcnt data mover path, non-temporal) so no VGPR landing space is
// needed; wave waits on s_wait_asynccnt, sums from LDS (compiler keeps the
// 16 values live for the square pass too), then writes 16 non-temporal b128
// stores. Both 256MB streams are touch-once -> NT hints keep L2 clean.

typedef __attribute__((ext_vector_type(4))) float v4f;
typedef int v4i __attribute__((vector_size(16)));   // matches builtin's param

#define NFRAMES 16
#define VEC4    1048576   // (4*64*128*128)/4 float4 elements per frame
#define TPB     256
#define CPOL_TH_NT 1      // gfx12 CPol: TH[2:0]=1 (non-temporal), scope WGP

#if defined(__AMDGCN__) &&                                             \
    __has_builtin(__builtin_amdgcn_global_load_async_to_lds_b128) &&   \
    __has_builtin(__builtin_amdgcn_s_wait_asynccnt)
#define USE_ASYNC_LDS 1
#else
#define USE_ASYNC_LDS 0
#endif

#if USE_ASYNC_LDS
typedef __attribute__((address_space(1))) v4i g_v4i;   // global int4*
typedef __attribute__((address_space(3))) v4i l_v4i;   // LDS int4*
#endif

__global__ __launch_bounds__(TPB) void frame_sim_kernel(
    const float* __restrict__ in, float* __restrict__ out)
{
    const size_t v = (size_t)blockIdx.x * TPB + threadIdx.x;  // float4 index
    const v4f* __restrict__ pin  = (const v4f*)in;
    v4f* __restrict__       pout = (v4f*)out;

#if USE_ASYNC_LDS
    // Each lane copies its own 16 frame values global->LDS via the async
    // engine; lanes consume only their own slots, so s_wait_asynccnt alone
    // orders producer->consumer (no workgroup barrier needed).
    __shared__ v4f tile[NFRAMES][TPB];   // 64 KB of the 320 KB WGP LDS
    const unsigned tid = threadIdx.x;
#pragma unroll
    for (int n = 0; n < NFRAMES; ++n) {
        __builtin_amdgcn_global_load_async_to_lds_b128(
            (g_v4i*)(uintptr_t)(pin + (size_t)n * VEC4 + v),
            (l_v4i*)(uintptr_t)&tile[n][tid],
            /*offset=*/0, /*cpol=*/CPOL_TH_NT);
    }
    __builtin_amdgcn_s_wait_asynccnt(0);
    asm volatile("" ::: "memory");   // keep ds_loads after the wait

    v4f s = tile[0][tid];
#pragma unroll
    for (int n = 1; n < NFRAMES; ++n) s += tile[n][tid];

#pragma unroll
    for (int n = 0; n < NFRAMES; ++n) {
        v4f f = tile[n][tid];
        __builtin_nontemporal_store(f * f * s, pout + (size_t)n * VEC4 + v);
    }
#else
    // Fallback: 16 non-temporal b128 loads in flight per thread.
    v4f f[NFRAMES];
#pragma unroll
    for (int n = 0; n < NFRAMES; ++n)
        f[n] = __builtin_nontemporal_load(pin + (size_t)n * VEC4 + v);

    v4f s = f[0];
#pragma unroll
    for (int n = 1; n < NFRAMES; ++n) s += f[n];

#pragma unroll
    for (int n = 0; n < NFRAMES; ++n)
        __builtin_nontemporal_store(f[n] * f[n] * s,
                                    pout + (size_t)n * VEC4 + v);
#endif
}

extern "C" void kernel_launch(void* const* d_in, const int* in_sizes, int n_in,
                              void* d_out, int out_size, void* d_ws, size_t ws_size,
                              hipStream_t stream)
{
    const float* frames = (const float*)d_in[0];
    float*       out    = (float*)d_out;

    dim3 grid(VEC4 / TPB);  // 4096 blocks x 256 threads = 1M threads,
    dim3 block(TPB);        // each covering 4 spatial f32 x 16 frames.
    frame_sim_kernel<<<grid, block, 0, stream>>>(frames, out);
}